// HardDetectionModule_4329327034802
// MI455X (gfx1250) — compile-verified
//
#include <hip/hip_runtime.h>

#define BN 2
#define CN 512
#define HN 192
#define WN 256

#define TILE_H 32
#define TILE_W 128
#define LDS_W  136   // 4-float left skirt + 128 + 4-float right skirt (16B chunks)
#define LDS_H  34    // 1 halo row above + 32 + 1 halo row below
#define NTHREADS 512

// ---------------------------------------------------------------------------
// Kernel 1: depth-wise (channel) max.  blockIdx.y = batch (no int division).
// Each thread reduces 512 channels for 4 consecutive pixels with float4 loads
// (32 lanes x 16B = fully coalesced, 512B/wave per channel step).
// ---------------------------------------------------------------------------
__global__ void __launch_bounds__(256)
dwmax_kernel(const float* __restrict__ src, float* __restrict__ dwmax) {
    const int b   = blockIdx.y;
    const int t   = blockIdx.x * blockDim.x + threadIdx.x;    // 12288 per batch
    const int stride4 = (HN * WN) / 4;                        // plane stride (float4)

    const float4* p = reinterpret_cast<const float4*>(src)
                    + (size_t)b * CN * stride4 + t;

    float4 m = p[0];
#pragma unroll 8
    for (int c = 1; c < CN; ++c) {
        float4 v = p[(size_t)c * stride4];
        m.x = fmaxf(m.x, v.x);
        m.y = fmaxf(m.y, v.y);
        m.z = fmaxf(m.z, v.z);
        m.w = fmaxf(m.w, v.w);
    }
    reinterpret_cast<float4*>(dwmax)[(size_t)b * stride4 + t] = m;
}

// ---------------------------------------------------------------------------
// Kernel 2: stage a zero-padded 34x136 halo tile into LDS with CDNA5 async
// global->LDS b128 copies (ASYNCcnt), then evaluate 3x3 local-max (-inf pad),
// Hessian edge test (zero pad = LDS skirt zeros), and depth-max compare.
// ---------------------------------------------------------------------------
__global__ void __launch_bounds__(NTHREADS)
detect_kernel(const float* __restrict__ src,
              const float* __restrict__ dwmax,
              float* __restrict__ out) {
    __shared__ float tile[LDS_H * LDS_W];

    const int tiles_x = WN / TILE_W;                 // 2
    const int tj  = blockIdx.x % tiles_x;
    const int ti  = blockIdx.x / tiles_x;            // 0..5
    const int ch  = blockIdx.y;
    const int b   = blockIdx.z;

    const int gi0 = ti * TILE_H;
    const int gj0 = tj * TILE_W;

    const size_t plane = (size_t)HN * WN;
    const float* splane = src + ((size_t)b * CN + ch) * plane;

    const int tid = threadIdx.x;

    // ---- stage halo tile: 34 rows x 34 b128 chunks = 1156 async copies -----
    const int CPR    = LDS_W / 4;        // 34 chunks per row
    const int NCHUNK = LDS_H * CPR;      // 1156
    for (int t = tid; t < NCHUNK; t += NTHREADS) {
        const int r    = t / CPR;
        const int c4   = t - r * CPR;
        const int grow = gi0 - 1 + r;
        const int gcol = gj0 - 4 + c4 * 4;           // multiple of 4, 16B aligned
        float* lp = &tile[r * LDS_W + c4 * 4];
        const bool valid = (grow >= 0) & (grow < HN) & (gcol >= 0) & (gcol < WN);
        if (valid) {
            unsigned lds_off = (unsigned)(uintptr_t)lp;   // low 32 bits = LDS offset
            unsigned long long ga =
                (unsigned long long)(uintptr_t)(splane + (size_t)grow * WN + gcol);
            asm volatile("global_load_async_to_lds_b128 %0, %1, off"
                         :: "v"(lds_off), "v"(ga) : "memory");
        } else {
            *reinterpret_cast<float4*>(lp) = make_float4(0.f, 0.f, 0.f, 0.f);
        }
    }
    asm volatile("s_wait_asynccnt 0" ::: "memory");
    __syncthreads();

    // ---- compute 32x128 tile, 8 elements per thread ------------------------
    const float thr  = 7.2f;                         // (5+1)^2 / 5
    const float NINF = -__builtin_inff();
    const float* dwp = dwmax + (size_t)b * plane;
    float* oplane    = out + ((size_t)b * CN + ch) * plane;

    const int lj  = tid & (TILE_W - 1);              // 0..127
    const int li0 = tid >> 7;                        // 0..3
#pragma unroll
    for (int k = 0; k < 8; ++k) {
        const int li = li0 + 4 * k;                  // 0..31
        const int gi = gi0 + li;
        const int gj = gj0 + lj;
        const int ci = (li + 1) * LDS_W + (lj + 4);

        const float v  = tile[ci];
        const float up = tile[ci - LDS_W];
        const float dn = tile[ci + LDS_W];
        const float lf = tile[ci - 1];
        const float rt = tile[ci + 1];
        const float ul = tile[ci - LDS_W - 1];
        const float ur = tile[ci - LDS_W + 1];
        const float dl = tile[ci + LDS_W - 1];
        const float dr = tile[ci + LDS_W + 1];

        // 3x3 local max with -inf semantics outside the image
        const bool iu = (gi > 0), id = (gi < HN - 1);
        const bool jl = (gj > 0), jr = (gj < WN - 1);
        float m = v;
        m = fmaxf(m, iu         ? up : NINF);
        m = fmaxf(m, id         ? dn : NINF);
        m = fmaxf(m, jl         ? lf : NINF);
        m = fmaxf(m, jr         ? rt : NINF);
        m = fmaxf(m, (iu && jl) ? ul : NINF);
        m = fmaxf(m, (iu && jr) ? ur : NINF);
        m = fmaxf(m, (id && jl) ? dl : NINF);
        m = fmaxf(m, (id && jr) ? dr : NINF);
        const bool is_lmax = (v == m);

        // Hessian stencil: LDS skirt holds zeros, matching zero-pad semantics
        const float dii = up + dn - 2.0f * v;
        const float djj = lf + rt - 2.0f * v;
        const float dij = 0.25f * (ul - ur - dl + dr);
        const float det = dii * djj - dij * dij;
        const float trc = dii + djj;
        const bool not_edge = ((trc * trc) / det <= thr) && (det > 0.0f);

        const bool is_dmax = (v == dwp[(size_t)gi * WN + gj]);

        const float res = (is_dmax && is_lmax && not_edge) ? 1.0f : 0.0f;
        __builtin_nontemporal_store(res, &oplane[(size_t)gi * WN + gj]);
    }
}

// ---------------------------------------------------------------------------
extern "C" void kernel_launch(void* const* d_in, const int* in_sizes, int n_in,
                              void* d_out, int out_size, void* d_ws, size_t ws_size,
                              hipStream_t stream) {
    (void)in_sizes; (void)n_in; (void)out_size; (void)ws_size;
    const float* src = (const float*)d_in[0];
    float* out       = (float*)d_out;
    float* dwmax     = (float*)d_ws;            // needs B*H*W*4 = 384 KiB

    // Pass 1: channel max -> workspace.  grid (48, 2): 48 blocks per batch.
    dim3 g1((HN * WN / 4) / 256, BN);
    dwmax_kernel<<<g1, 256, 0, stream>>>(src, dwmax);

    // Pass 2: detection (one block per batch/channel/32x128 tile)
    dim3 g2((HN / TILE_H) * (WN / TILE_W), CN, BN);   // (12, 512, 2)
    detect_kernel<<<g2, NTHREADS, 0, stream>>>(src, dwmax, out);
}